// STSSIM_26096221290789
// MI455X (gfx1250) — compile-verified
//
#include <hip/hip_runtime.h>

typedef __attribute__((ext_vector_type(2))) float v2f;
typedef __attribute__((ext_vector_type(8))) float v8f;

#define IMG      512
#define HWPLANE  (512 * 512)
#define PROWS    32        // 16-row band + 10 halo, padded to 32 (rows >=26 weight-zero)
#define PSTRIDE  140       // 128 + 10 halo, padded even for float2 LDS loads
#define HSTRIDE  17
#define HROWS    32

// 11-tap normalized Gaussian, sigma = 1.5
__constant__ float G11[11] = {
    0.00102838f, 0.00759876f, 0.03600080f, 0.10936070f, 0.21300553f,
    0.26601243f, 0.21300553f, 0.10936070f, 0.03600080f, 0.00759876f,
    0.00102838f};

template <int Q>
__device__ __forceinline__ float qval(float xv, float yv) {
  if constexpr (Q == 0) return xv;
  if constexpr (Q == 1) return yv;
  if constexpr (Q == 2) return xv * xv;
  if constexpr (Q == 3) return yv * yv;
  return xv * yv;
}

// vertical pass: Out(16x16) = G(16x26) x H(26x16); H bounced through LDS to
// convert the C/D register layout into B fragments. Rows >= 26 of H are
// multiplied by zero weights, so their (finite) contents are irrelevant.
__device__ __forceinline__ v8f stssim_vertical(float* Hl, const v8f hT, const v8f hB,
                                               const v2f* wfrag, int half, int nm) {
#pragma unroll
  for (int r = 0; r < 8; ++r) {
    Hl[(r + 8 * half) * HSTRIDE + nm]      = hT[r];
    Hl[(16 + r + 8 * half) * HSTRIDE + nm] = hB[r];
  }
  __syncthreads();
  v8f acc = {};
#pragma unroll
  for (int ch = 0; ch < 7; ++ch) {
    const int kb = ch * 4 + 2 * half;
    v2f b;
    b[0] = Hl[kb * HSTRIDE + nm];
    b[1] = Hl[(kb + 1) * HSTRIDE + nm];
    acc = __builtin_amdgcn_wmma_f32_16x16x4_f32(false, wfrag[ch], false, b,
                                                (short)0, acc, false, false);
  }
  __syncthreads();
  return acc;
}

// one pass computes quantity QA (and optionally QB) for this wave's tile:
// horizontal H(26x16) = P(26x26) x W(26x16) in 7 K-chunks x 2 row-groups,
// then vertical via stssim_vertical.
template <int QA, int QB>
__device__ __forceinline__ void stssim_pass(const float* xp, const float* yp,
                                            float* Hl, const v2f* wfrag,
                                            int half, int nm, int cbase,
                                            v8f& outA, v8f& outB) {
  v8f hTa = {}, hBa = {}, hTb = {}, hBb = {};
  const float* rT = xp + nm * PSTRIDE + cbase;          // rows 0..15
  const float* rB = xp + (16 + nm) * PSTRIDE + cbase;   // rows 16..31 (26+ weight-zero)
  const long long ydelta = yp - xp;
#pragma unroll
  for (int ch = 0; ch < 7; ++ch) {
    const int k0 = ch * 4 + 2 * half;                   // even -> 8B aligned
    const float2 xT = *(const float2*)(rT + k0);
    const float2 yT = *(const float2*)(rT + ydelta + k0);
    const float2 xB = *(const float2*)(rB + k0);
    const float2 yB = *(const float2*)(rB + ydelta + k0);
    const v2f b = wfrag[ch];
    v2f a;
    a[0] = qval<QA>(xT.x, yT.x);  a[1] = qval<QA>(xT.y, yT.y);
    hTa = __builtin_amdgcn_wmma_f32_16x16x4_f32(false, a, false, b, (short)0, hTa, false, false);
    a[0] = qval<QA>(xB.x, yB.x);  a[1] = qval<QA>(xB.y, yB.y);
    hBa = __builtin_amdgcn_wmma_f32_16x16x4_f32(false, a, false, b, (short)0, hBa, false, false);
    if constexpr (QB >= 0) {
      a[0] = qval<QB>(xT.x, yT.x);  a[1] = qval<QB>(xT.y, yT.y);
      hTb = __builtin_amdgcn_wmma_f32_16x16x4_f32(false, a, false, b, (short)0, hTb, false, false);
      a[0] = qval<QB>(xB.x, yB.x);  a[1] = qval<QB>(xB.y, yB.y);
      hBb = __builtin_amdgcn_wmma_f32_16x16x4_f32(false, a, false, b, (short)0, hBb, false, false);
    }
  }
  outA = stssim_vertical(Hl, hTa, hBa, wfrag, half, nm);
  if constexpr (QB >= 0)
    outB = stssim_vertical(Hl, hTb, hBb, wfrag, half, nm);
}

// ---------------------------------------------------------------------------
// One block = 16-row x 128-col strip of one (frame,channel) plane.
// 8 waves, each wave computes one 16x16 SSIM-map tile via WMMA f32 16x16x4.
// mode 0: spatial SSIM on x,y.  mode 1: temporal SSIM on frame diffs.
// ---------------------------------------------------------------------------
__global__ __launch_bounds__(256)
void stssim_tiles(const float* __restrict__ x, const float* __restrict__ y,
                  float* __restrict__ accum, int mode) {
  __shared__ float xp[PROWS * PSTRIDE];
  __shared__ float yp[PROWS * PSTRIDE];
  __shared__ float Hbuf[8][HROWS * HSTRIDE];   // per-wave H scratch
  __shared__ float Glds[11];
  __shared__ float wavesum[8];

  const int tid  = threadIdx.x;
  const int lane = tid & 31;
  const int wave = tid >> 5;

  // ---- plane decode --------------------------------------------------------
  long long base0, base1 = 0;
  const int plane = blockIdx.y;
  if (mode == 0) {
    base0 = (long long)plane * HWPLANE;              // (b*T+t)*C + c == plane
  } else {
    const int b = plane / 45, rem = plane % 45;      // 45 = (T-1)*C
    const int t = rem / 3, c = rem % 3;
    base0 = (long long)((b * 16 + t) * 3 + c) * HWPLANE;
    base1 = (long long)((b * 16 + t + 1) * 3 + c) * HWPLANE;
  }
  const int tileRow = (int)blockIdx.x >> 2;          // 0..31
  const int colBlk  = (int)blockIdx.x & 3;           // 0..3
  const int row0 = tileRow * 16 - 5;
  const int col0 = colBlk * 128 - 5;

  if (tid < 11) Glds[tid] = G11[tid];

  // ---- stage zero-padded x / y patches in LDS (no divergence downstream) ---
  for (int idx = tid; idx < PROWS * PSTRIDE; idx += 256) {
    const int pr = idx / PSTRIDE, pc = idx % PSTRIDE;
    float vx = 0.f, vy = 0.f;
    const int r = row0 + pr, c = col0 + pc;
    if ((unsigned)r < IMG && (unsigned)c < IMG) {
      const long long off = (long long)r * IMG + c;
      if (mode == 0) {
        vx = x[base0 + off];
        vy = y[base0 + off];
      } else {
        vx = x[base1 + off] - x[base0 + off];
        vy = y[base1 + off] - y[base0 + off];
      }
    }
    xp[idx] = vx;
    yp[idx] = vy;
  }
  __syncthreads();

  // ---- per-lane constant Gaussian fragments (branch-free LDS gather) -------
  // Serves as B (W[m][c]=g[m-c]) in horizontal pass AND A (G[r][m]=g[m-r])
  // in vertical pass (identical per-lane values under the ISA layouts).
  // The empty asm pins each value in a VGPR so the register allocator cannot
  // rematerialize the gather + select chain inside the WMMA loops.
  const int half  = lane >> 4;       // 0: lanes 0-15, 1: lanes 16-31
  const int nm    = lane & 15;       // N (or M) index of this lane
  const int cbase = wave * 16;       // this wave's tile offset in the patch
  v2f wfrag[7];
#pragma unroll
  for (int ch = 0; ch < 7; ++ch) {
    const int kb = ch * 4 + 2 * half;
    const int d0 = kb - nm, d1 = d0 + 1;
    const int i0 = d0 < 0 ? 0 : (d0 > 10 ? 10 : d0);
    const int i1 = d1 < 0 ? 0 : (d1 > 10 ? 10 : d1);
    float w0 = ((unsigned)d0 <= 10u) ? Glds[i0] : 0.f;
    float w1 = ((unsigned)d1 <= 10u) ? Glds[i1] : 0.f;
    asm volatile("" : "+v"(w0), "+v"(w1));   // pin: no remat inside hot loops
    v2f t;
    t[0] = w0;
    t[1] = w1;
    wfrag[ch] = t;
  }

  // ---- three low-pressure passes over the five blurred quantities ----------
  float* Hl = &Hbuf[wave][0];
  v8f o_mu1, o_mu2, o_xx, o_yy, o_xy, o_dummy;
  stssim_pass<0, 2>(xp, yp, Hl, wfrag, half, nm, cbase, o_mu1, o_xx);
  stssim_pass<1, 3>(xp, yp, Hl, wfrag, half, nm, cbase, o_mu2, o_yy);
  stssim_pass<4, -1>(xp, yp, Hl, wfrag, half, nm, cbase, o_xy, o_dummy);

  // ---- elementwise SSIM + reduction ---------------------------------------
  float s = 0.f;
#pragma unroll
  for (int r = 0; r < 8; ++r) {
    const float mu1 = o_mu1[r], mu2 = o_mu2[r];
    const float mu1sq = mu1 * mu1, mu2sq = mu2 * mu2, mu12 = mu1 * mu2;
    const float s1 = o_xx[r] - mu1sq, s2 = o_yy[r] - mu2sq, s12 = o_xy[r] - mu12;
    const float num = (2.f * mu12 + 1.0e-4f) * (2.f * s12 + 9.0e-4f);
    const float den = (mu1sq + mu2sq + 1.0e-4f) * (s1 + s2 + 9.0e-4f);
    s += num / den;
  }
  for (int off = 16; off > 0; off >>= 1) s += __shfl_xor(s, off, 32);
  if (lane == 0) wavesum[wave] = s;
  __syncthreads();
  if (tid == 0) {
    float bs = 0.f;
#pragma unroll
    for (int w = 0; w < 8; ++w) bs += wavesum[w];
    atomicAdd(&accum[mode], bs);
  }
}

__global__ void stssim_init(float* a) {
  if (threadIdx.x < 2) a[threadIdx.x] = 0.f;
}

__global__ void stssim_finalize(const float* a, float* out) {
  const float sm = a[0] * (1.0f / 50331648.0f);   // 64*3*512*512
  const float tm = a[1] * (1.0f / 47185920.0f);   // 60*3*512*512
  out[0] = 0.5f * (1.0f - sm) + 0.5f * (1.0f - tm);
}

extern "C" void kernel_launch(void* const* d_in, const int* in_sizes, int n_in,
                              void* d_out, int out_size, void* d_ws, size_t ws_size,
                              hipStream_t stream) {
  const float* x = (const float*)d_in[0];
  const float* y = (const float*)d_in[1];
  float* accum = (float*)d_ws;
  float* out = (float*)d_out;

  stssim_init<<<1, 32, 0, stream>>>(accum);
  // spatial: 64 frames x 3 channels = 192 planes, 32x4 = 128 blocks/plane
  stssim_tiles<<<dim3(128, 192), dim3(256), 0, stream>>>(x, y, accum, 0);
  // temporal: 60 frame-pairs x 3 channels = 180 planes
  stssim_tiles<<<dim3(128, 180), dim3(256), 0, stream>>>(x, y, accum, 1);
  stssim_finalize<<<1, 1, 0, stream>>>(accum, out);
}